// HolographicPatternExtractor_56564719288965
// MI455X (gfx1250) — compile-verified
//
#include <hip/hip_runtime.h>
#include <hip/hip_bf16.h>

#define PI_F 3.14159265358979323846f

typedef __attribute__((ext_vector_type(2))) float v2f;
typedef __attribute__((ext_vector_type(8))) float v8f;

// ---------------------------------------------------------------------------
// gfx1250 async global->LDS path (ASYNCcnt). Guarded so the file compiles on
// toolchains lacking the builtins; fallback is plain load+store.
// Builtin signature (from hipcc diagnostic): (AS1 int*, AS3 int*, imm, imm).
// AS3 pointer = low 32 bits of the generic LDS address (ISA 10.2 aperture).
// ---------------------------------------------------------------------------
#if __has_builtin(__builtin_amdgcn_global_load_async_to_lds_b32) && \
    __has_builtin(__builtin_amdgcn_s_wait_asynccnt)
#define USE_ASYNC_LDS 1
typedef __attribute__((address_space(3))) int lds_int;
typedef __attribute__((address_space(1))) int glb_int;
static __device__ __forceinline__ void async_g2l_b32(const void* g, void* l) {
  __builtin_amdgcn_global_load_async_to_lds_b32(
      (glb_int*)(unsigned long long)g,
      (lds_int*)(unsigned int)(unsigned long long)l, 0, 0);
}
static __device__ __forceinline__ void async_wait0() {
  __builtin_amdgcn_s_wait_asynccnt(0);
}
#endif

// ---------------------------------------------------------------------------
// WMMA helper: D = A(16x4 f32) * B(4x16 f32) + C(16x16 f32), wave32.
// A: lane m=lane&15 holds K=(lane>>4)*2 .. +1 in .x/.y
// B: lane n=lane&15 holds K=(lane>>4)*2 .. +1 in .x/.y
// C/D: element r of v8f is (m = r + (lane>>4)*8, n = lane&15)
// ---------------------------------------------------------------------------
static __device__ __forceinline__ v8f wmma4(v2f a, v2f b, v8f c) {
  return __builtin_amdgcn_wmma_f32_16x16x4_f32(false, a, false, b, (short)0, c,
                                               false, false);
}

// ---------------------------------------------------------------------------
// 512-point radix-2 DIT FFT in LDS. 256 threads, input already bit-reversed.
// ---------------------------------------------------------------------------
static __device__ __forceinline__ void fft512(float* sre, float* sim_, int t) {
  for (int len = 2; len <= 512; len <<= 1) {
    int half = len >> 1;
    __syncthreads();
    int j = t & (half - 1);
    int blk = t / half;
    int i1 = blk * len + j;
    int i2 = i1 + half;
    float ang = -2.0f * PI_F * (float)j / (float)len;
    float wr = __cosf(ang);
    float wi = __sinf(ang);
    float xr = sre[i2], xi = sim_[i2];
    float tr = wr * xr - wi * xi;
    float ti = wr * xi + wi * xr;
    float ur = sre[i1], ui = sim_[i1];
    sre[i1] = ur + tr; sim_[i1] = ui + ti;
    sre[i2] = ur - tr; sim_[i2] = ui - ti;
  }
  __syncthreads();
}

// ---------------- zero scratch region -------------------------------------
__global__ void k_zero(float* p, int n) {
  int i = blockIdx.x * blockDim.x + threadIdx.x;
  if (i < n) p[i] = 0.0f;
}

// ---------------- FFT over rows (real input) -------------------------------
__global__ void k_fft_rows(const float* __restrict__ img,
                           float* __restrict__ bre, float* __restrict__ bim) {
  __shared__ float sre[512], sim_[512];
  int row = blockIdx.x;                 // b*512 + h
  int t = threadIdx.x;                  // 256
  const float* src = img + (size_t)row * 512;
  for (int e = t; e < 512; e += 256) {
    int r = (int)(__brev((unsigned)e) >> 23);
#ifdef USE_ASYNC_LDS
    async_g2l_b32(&src[e], &sre[r]);    // lands at bit-reversed position
#else
    sre[r] = src[e];
#endif
    sim_[r] = 0.0f;
  }
#ifdef USE_ASYNC_LDS
  async_wait0();                        // per-wave; barrier in fft512 publishes
#endif
  fft512(sre, sim_, t);
  float* dre = bre + (size_t)row * 512;
  float* dim = bim + (size_t)row * 512;
  for (int e = t; e < 512; e += 256) { dre[e] = sre[e]; dim[e] = sim_[e]; }
}

// ---------------- FFT over cols + magnitude + phase statistics -------------
__global__ void k_fft_cols(const float* __restrict__ bre,
                           const float* __restrict__ bim,
                           float* __restrict__ mag,
                           float* psum, float* psumsq, float* hist) {
  __shared__ float sre[512], sim_[512];
  __shared__ float lh[64];
  __shared__ float lsum[2];
  int b = blockIdx.x >> 9;
  int w = blockIdx.x & 511;
  int t = threadIdx.x;                  // 256
  size_t base = (((size_t)b) << 18) + (size_t)w;
  for (int e = t; e < 512; e += 256) {
    int r = (int)(__brev((unsigned)e) >> 23);
#ifdef USE_ASYNC_LDS
    async_g2l_b32(&bre[base + ((size_t)e << 9)], &sre[r]);
    async_g2l_b32(&bim[base + ((size_t)e << 9)], &sim_[r]);
#else
    sre[r]  = bre[base + ((size_t)e << 9)];
    sim_[r] = bim[base + ((size_t)e << 9)];
#endif
  }
  if (t < 64) lh[t] = 0.0f;
  if (t < 2) lsum[t] = 0.0f;
#ifdef USE_ASYNC_LDS
  async_wait0();
#endif
  fft512(sre, sim_, t);                 // first internal sync covers init
  float s1 = 0.0f, s2 = 0.0f;
  for (int e = t; e < 512; e += 256) {
    float re = sre[e], im = sim_[e];
    float mg = sqrtf(re * re + im * im);
    mag[(((size_t)b) << 18) + ((size_t)e << 9) + w] = mg;
    float ph = atan2f(im, re);          // jnp.angle
    s1 += ph; s2 += ph * ph;
    int bin = (int)floorf((ph + PI_F) * (64.0f / (2.0f * PI_F)));
    bin = bin < 0 ? 0 : (bin > 63 ? 63 : bin);
    atomicAdd(&lh[bin], 1.0f);
  }
  atomicAdd(&lsum[0], s1);
  atomicAdd(&lsum[1], s2);
  __syncthreads();
  if (t < 64) atomicAdd(&hist[b * 64 + t], lh[t]);
  if (t == 0) { atomicAdd(&psum[b], lsum[0]); atomicAdd(&psumsq[b], lsum[1]); }
}

// ---------------- phase features: mean, std(ddof=1), histogram entropy -----
__global__ void k_phase_feats(const float* psum, const float* psumsq,
                              const float* hist, float* out_phase,
                              float* combined) {
  __shared__ float sh[64];
  int b = blockIdx.x;                   // 16 blocks x 64 threads
  int t = threadIdx.x;
  const float Nf = 262144.0f;
  float p = hist[b * 64 + t] / Nf;
  sh[t] = -p * logf(p + 1e-10f);
  __syncthreads();
  for (int s = 32; s > 0; s >>= 1) {
    if (t < s) sh[t] += sh[t + s];
    __syncthreads();
  }
  if (t == 0) {
    float mean = psum[b] / Nf;
    float var = (psumsq[b] - Nf * mean * mean) / (Nf - 1.0f);
    float sd = sqrtf(fmaxf(var, 0.0f));
    float ent = sh[0];
    out_phase[b * 3 + 0] = mean;
    out_phase[b * 3 + 1] = sd;
    out_phase[b * 3 + 2] = ent;
    combined[b * 1120 + 1088 + 0] = mean;
    combined[b * 1120 + 1088 + 1] = sd;
    combined[b * 1120 + 1088 + 2] = ent;
  }
}

// ---------------- top-128 (iterative argmax, destructive) + L2 normalize ---
__global__ void k_topk(float* __restrict__ mag, float* __restrict__ domN) {
  __shared__ float smax[1024];
  __shared__ int   sidx[1024];
  __shared__ float dom_s[128];
  int b = blockIdx.x;
  int t = threadIdx.x;                  // 1024
  float* m = mag + (((size_t)b) << 18);
  for (int iter = 0; iter < 128; ++iter) {
    float best = -2.0f; int bi = 0;
    for (int i = t; i < 262144; i += 1024) {
      float v = m[i];
      if (v > best) { best = v; bi = i; }
    }
    smax[t] = best; sidx[t] = bi;
    __syncthreads();
    for (int s = 512; s > 0; s >>= 1) {
      if (t < s && smax[t + s] > smax[t]) { smax[t] = smax[t + s]; sidx[t] = sidx[t + s]; }
      __syncthreads();
    }
    if (t == 0) { dom_s[iter] = smax[0]; m[sidx[0]] = -1.0f; }
    __syncthreads();
  }
  if (t < 128) smax[t] = dom_s[t] * dom_s[t];
  __syncthreads();
  for (int s = 64; s > 0; s >>= 1) {
    if (t < s && t + s < 128) smax[t] += smax[t + s];
    __syncthreads();
  }
  float nrm = fmaxf(sqrtf(smax[0]), 1e-12f);
  if (t < 128) domN[b * 128 + t] = dom_s[t] / nrm;
}

// ---------------- receptor row L2 normalize --------------------------------
__global__ void k_recnorm(const float* __restrict__ rec, float* __restrict__ recN) {
  __shared__ float sh[128];
  int r = blockIdx.x;                   // 64 blocks x 128 threads
  int t = threadIdx.x;
  float v = rec[r * 128 + t];
  sh[t] = v * v;
  __syncthreads();
  for (int s = 64; s > 0; s >>= 1) {
    if (t < s) sh[t] += sh[t + s];
    __syncthreads();
  }
  float nrm = fmaxf(sqrtf(sh[0]), 1e-12f);
  recN[r * 128 + t] = v / nrm;
}

// ---------------- sims = domN[16,128] x recN^T[128,64] via WMMA f32 --------
__global__ void k_sims(const float* __restrict__ domN, const float* __restrict__ recN,
                       float* __restrict__ out_sims, float* __restrict__ combined) {
  int nb = blockIdx.x * 16;             // 4 blocks x 32 threads (1 wave)
  int lane = threadIdx.x;
  int mn = lane & 15;
  int koff = (lane >> 4) * 2;
  int half = lane >> 4;
  v8f c = {};
  for (int kb = 0; kb < 128; kb += 4) {
    v2f a, bb;
    a.x = domN[mn * 128 + kb + koff];
    a.y = domN[mn * 128 + kb + koff + 1];
    bb.x = recN[(nb + mn) * 128 + kb + koff];     // B[k][n] = recN[n][k]
    bb.y = recN[(nb + mn) * 128 + kb + koff + 1];
    c = wmma4(a, bb, c);
  }
#pragma unroll
  for (int r = 0; r < 8; ++r) {
    int m = r + half * 8;
    int n = nb + mn;
    float v = c[r];
    out_sims[m * 64 + n] = v;
    combined[m * 1120 + n] = v;
  }
}

// ---------------- gradients (jnp.gradient semantics) -----------------------
__global__ void k_grad(const float* __restrict__ img,
                       float* __restrict__ gmag, float* __restrict__ gori) {
  int idx = blockIdx.x * blockDim.x + threadIdx.x;
  if (idx >= 16 * 512 * 512) return;
  int w = idx & 511;
  int h = (idx >> 9) & 511;
  int b = idx >> 18;
  const float* p = img + (((size_t)b) << 18);
  float c0 = p[(h << 9) + w];
  float gy, gx;
  if (h == 0)        gy = p[(1 << 9) + w] - c0;
  else if (h == 511) gy = c0 - p[(510 << 9) + w];
  else               gy = 0.5f * (p[((h + 1) << 9) + w] - p[((h - 1) << 9) + w]);
  if (w == 0)        gx = p[(h << 9) + 1] - c0;
  else if (w == 511) gx = c0 - p[(h << 9) + 510];
  else               gx = 0.5f * (p[(h << 9) + w + 1] - p[(h << 9) + w - 1]);
  gmag[idx] = sqrtf(gx * gx + gy * gy);
  gori[idx] = atan2f(gy, gx);
}

// ---------------- conv1 (2->32, 3x3, SAME) + ReLU + maxpool 2x2 ------------
__global__ void k_conv1_pool(const float* __restrict__ g0, const float* __restrict__ g1,
                             const float* __restrict__ w1, const float* __restrict__ b1,
                             float* __restrict__ pool1) {
  int idx = blockIdx.x * blockDim.x + threadIdx.x;
  if (idx >= 16 * 32 * 256 * 256) return;
  int px = idx & 255;
  int py = (idx >> 8) & 255;
  int oc = (idx >> 16) & 31;
  int b  = idx >> 21;
  const float* in0 = g0 + (((size_t)b) << 18);
  const float* in1 = g1 + (((size_t)b) << 18);
  const float* wt = w1 + oc * 18;       // [32][2][3][3]
  float bias = b1[oc];
  float best = 0.0f;                    // relu outputs are >= 0
  for (int dy = 0; dy < 2; ++dy)
    for (int dx = 0; dx < 2; ++dx) {
      int y = py * 2 + dy, x = px * 2 + dx;
      float acc = bias;
#pragma unroll
      for (int ky = 0; ky < 3; ++ky) {
        int yy = y + ky - 1;
        if (yy < 0 || yy > 511) continue;
#pragma unroll
        for (int kx = 0; kx < 3; ++kx) {
          int xx = x + kx - 1;
          if (xx < 0 || xx > 511) continue;
          acc += wt[ky * 3 + kx]     * in0[(yy << 9) + xx]
               + wt[9 + ky * 3 + kx] * in1[(yy << 9) + xx];
        }
      }
      best = fmaxf(best, fmaxf(acc, 0.0f));
    }
  pool1[idx] = best;
}

// ---------------------------------------------------------------------------
// conv2 implicit-GEMM WMMA + ReLU + 4x4 adaptive avg.
// K permuted as k = tap*32 + ic (same permutation on A and B -> GEMM invariant)
// so the 9-tap logic hoists out of the inner loop. Per-wave input halo tile
// (32ch x 3 rows x 18 cols) staged in LDS via async global->LDS loads; the
// region is wave-private, so a per-wave s_wait_asynccnt(0) suffices (no bar).
// ---------------------------------------------------------------------------
__global__ void k_conv2_avg(const float* __restrict__ pool1,
                            const float* __restrict__ w2, const float* __restrict__ b2,
                            float* __restrict__ out_spatial, float* __restrict__ combined) {
  __shared__ float stage[8 * 1728];     // [wave][ic=32][row=3][col=18]
  __shared__ float csum[64];
  int blk = blockIdx.x;                 // 256 blocks (16 b x 16 regions)
  int b = blk >> 4;
  int region = blk & 15;
  int ry = region >> 2, rx = region & 3;
  int t = threadIdx.x;                  // 256 threads = 8 waves
  int wave = t >> 5, lane = t & 31;
  if (t < 64) csum[t] = 0.0f;
  __syncthreads();
  int mn = lane & 15;
  int koff = (lane >> 4) * 2;
  int wbase = wave * 1728;
  const float* in = pool1 + ((size_t)b) * (32u * 256u * 256u);
  float lsum0 = 0.f, lsum1 = 0.f, lsum2 = 0.f, lsum3 = 0.f;
  for (int g = wave; g < 256; g += 8) { // 16-pixel row strips of 64x64 region
    int py = ry * 64 + (g >> 2);
    int px0 = rx * 64 + (g & 3) * 16;
    // ---- stage input halo tile for this strip (wave-private LDS) ----
    for (int i = lane; i < 1728; i += 32) {
      int ic = i / 54;
      int rem = i - ic * 54;
      int row = rem / 18;
      int col = rem - row * 18;
      int yy = py - 1 + row;
      int xx = px0 - 1 + col;
      if (yy >= 0 && yy < 256 && xx >= 0 && xx < 256) {
#ifdef USE_ASYNC_LDS
        async_g2l_b32(&in[(((size_t)ic) << 16) + (yy << 8) + xx], &stage[wbase + i]);
#else
        stage[wbase + i] = in[(((size_t)ic) << 16) + (yy << 8) + xx];
#endif
      } else {
        stage[wbase + i] = 0.0f;
      }
    }
#ifdef USE_ASYNC_LDS
    async_wait0();
#endif
    // ---- 16x64 output strip: A from LDS, B from global (L2-resident) ----
    v8f a0 = {}, a1 = {}, a2 = {}, a3 = {};
#pragma unroll
    for (int r9 = 0; r9 < 9; ++r9) {
      int ky = r9 / 3;
      int kx = r9 - ky * 3;
      int sbase = wbase + ky * 18 + kx + mn;
      __builtin_prefetch(&w2[mn * 288 + r9], 0, 1);
      for (int icb = 0; icb < 32; icb += 4) {
        int ic0 = icb + koff;
        int si = sbase + ic0 * 54;
        int wk0 = ic0 * 9 + r9;         // permuted-k weight index
        v2f a, bb;
        a.x = stage[si];
        a.y = stage[si + 54];
        bb.x = w2[mn * 288 + wk0];        bb.y = w2[mn * 288 + wk0 + 9];
        a0 = wmma4(a, bb, a0);
        bb.x = w2[(16 + mn) * 288 + wk0]; bb.y = w2[(16 + mn) * 288 + wk0 + 9];
        a1 = wmma4(a, bb, a1);
        bb.x = w2[(32 + mn) * 288 + wk0]; bb.y = w2[(32 + mn) * 288 + wk0 + 9];
        a2 = wmma4(a, bb, a2);
        bb.x = w2[(48 + mn) * 288 + wk0]; bb.y = w2[(48 + mn) * 288 + wk0 + 9];
        a3 = wmma4(a, bb, a3);
      }
    }
#pragma unroll
    for (int r = 0; r < 8; ++r) {
      lsum0 += fmaxf(a0[r] + b2[mn],      0.0f);
      lsum1 += fmaxf(a1[r] + b2[16 + mn], 0.0f);
      lsum2 += fmaxf(a2[r] + b2[32 + mn], 0.0f);
      lsum3 += fmaxf(a3[r] + b2[48 + mn], 0.0f);
    }
  }
  atomicAdd(&csum[mn],      lsum0);
  atomicAdd(&csum[16 + mn], lsum1);
  atomicAdd(&csum[32 + mn], lsum2);
  atomicAdd(&csum[48 + mn], lsum3);
  __syncthreads();
  if (t < 64) {
    float avg = csum[t] * (1.0f / 4096.0f);
    out_spatial[b * 1024 + t * 16 + region] = avg;
    combined[b * 1120 + 64 + t * 16 + region] = avg;
  }
}

// ---------------- fc1: [16,1091] x [1091,512] + bias + ReLU (WMMA) ---------
__global__ void k_fc1(const float* __restrict__ combined,
                      const float* __restrict__ fw, const float* __restrict__ fb,
                      float* __restrict__ z1) {
  int nb = blockIdx.x * 16;             // 32 blocks x 32 threads
  int lane = threadIdx.x;
  int mn = lane & 15;
  int koff = (lane >> 4) * 2;
  int half = lane >> 4;
  v8f c = {};
  for (int kb = 0; kb < 1092; kb += 4) {
    int k0 = kb + koff, k1 = k0 + 1;
    v2f a, bb;
    a.x = combined[mn * 1120 + k0];     // padded cols are zero
    a.y = combined[mn * 1120 + k1];
    bb.x = (k0 < 1091) ? fw[(size_t)(nb + mn) * 1091 + k0] : 0.0f;
    bb.y = (k1 < 1091) ? fw[(size_t)(nb + mn) * 1091 + k1] : 0.0f;
    c = wmma4(a, bb, c);
  }
#pragma unroll
  for (int r = 0; r < 8; ++r) {
    int m = r + half * 8;
    int n = nb + mn;
    z1[m * 512 + n] = fmaxf(c[r] + fb[n], 0.0f);
  }
}

// ---------------- fc2: [16,512]x[512,128] + bias, then LayerNorm (WMMA) ----
__global__ void k_fc2_ln(const float* __restrict__ z1,
                         const float* __restrict__ fw, const float* __restrict__ fb,
                         const float* __restrict__ lg, const float* __restrict__ lb,
                         float* __restrict__ out_holo) {
  __shared__ float zsh[16 * 128];
  __shared__ float mu[16], rs[16];
  int t = threadIdx.x;                  // 256 = 8 waves; wave w owns n-tile w
  int wave = t >> 5, lane = t & 31;
  int mn = lane & 15;
  int koff = (lane >> 4) * 2;
  int half = lane >> 4;
  int nb = wave * 16;
  v8f c = {};
  for (int kb = 0; kb < 512; kb += 4) {
    int k0 = kb + koff;
    v2f a, bb;
    a.x = z1[mn * 512 + k0];
    a.y = z1[mn * 512 + k0 + 1];
    bb.x = fw[(nb + mn) * 512 + k0];
    bb.y = fw[(nb + mn) * 512 + k0 + 1];
    c = wmma4(a, bb, c);
  }
#pragma unroll
  for (int r = 0; r < 8; ++r) {
    int m = r + half * 8;
    int n = nb + mn;
    zsh[m * 128 + n] = c[r] + fb[n];
  }
  __syncthreads();
  if (t < 16) {
    float s = 0.0f;
    for (int i = 0; i < 128; ++i) s += zsh[t * 128 + i];
    float mean = s * (1.0f / 128.0f);
    float v = 0.0f;
    for (int i = 0; i < 128; ++i) { float d = zsh[t * 128 + i] - mean; v += d * d; }
    v *= (1.0f / 128.0f);               // jnp.var ddof=0
    mu[t] = mean;
    rs[t] = rsqrtf(v + 1e-5f);
  }
  __syncthreads();
  for (int i = t; i < 2048; i += 256) {
    int m = i >> 7, n = i & 127;
    out_holo[i] = (zsh[i] - mu[m]) * rs[m] * lg[n] + lb[n];
  }
}

// ===========================================================================
extern "C" void kernel_launch(void* const* d_in, const int* in_sizes, int n_in,
                              void* d_out, int out_size, void* d_ws, size_t ws_size,
                              hipStream_t stream) {
  const float* image = (const float*)d_in[0];
  const float* rec   = (const float*)d_in[1];
  const float* c1w   = (const float*)d_in[2];
  const float* c1b   = (const float*)d_in[3];
  const float* c2w   = (const float*)d_in[4];
  const float* c2b   = (const float*)d_in[5];
  const float* f1w   = (const float*)d_in[6];
  const float* f1b   = (const float*)d_in[7];
  const float* f2w   = (const float*)d_in[8];
  const float* f2b   = (const float*)d_in[9];
  const float* lng   = (const float*)d_in[10];
  const float* lnb   = (const float*)d_in[11];

  float* out = (float*)d_out;
  float* out_sims    = out;             // [16,64]
  float* out_spatial = out + 1024;      // [16,1024]
  float* out_phase   = out + 17408;     // [16,3]
  float* out_holo    = out + 17456;     // [16,128]

  float* ws = (float*)d_ws;
  const size_t NTOT = (size_t)16 * 512 * 512;           // 4194304
  float* bre      = ws;                                 // complex FFT re
  float* bim      = ws + NTOT;                          // complex FFT im
  float* mag      = ws + 2 * NTOT;                      // magnitudes (destroyed by topk)
  float* pool1    = ws + 3 * NTOT;                      // [16,32,256,256]
  float* domN     = pool1 + (size_t)16 * 32 * 256 * 256;
  float* recN     = domN + 16 * 128;
  float* zregion  = recN + 64 * 128;                    // contiguous zeroed region
  float* psum     = zregion;                            // 16
  float* psumsq   = zregion + 16;                       // 16
  float* hist     = zregion + 32;                       // 16*64
  float* combined = zregion + 32 + 1024;                // [16,1120] (padded)
  float* z1       = combined + 16 * 1120;               // [16,512]
  const int nzero = 32 + 1024 + 16 * 1120;              // 18976

  // 0) zero accumulators + padded combined buffer
  k_zero<<<(nzero + 255) / 256, 256, 0, stream>>>(zregion, nzero);
  // 1) FFT rows then cols (+ magnitude, phase stats)
  k_fft_rows<<<16 * 512, 256, 0, stream>>>(image, bre, bim);
  k_fft_cols<<<16 * 512, 256, 0, stream>>>(bre, bim, mag, psum, psumsq, hist);
  // 2) phase features
  k_phase_feats<<<16, 64, 0, stream>>>(psum, psumsq, hist, out_phase, combined);
  // 3) top-128 + normalize; receptor normalize; sims GEMM (WMMA)
  k_topk<<<16, 1024, 0, stream>>>(mag, domN);
  k_recnorm<<<64, 128, 0, stream>>>(rec, recN);
  k_sims<<<4, 32, 0, stream>>>(domN, recN, out_sims, combined);
  // 4) spatial path (gradients reuse the FFT complex buffers)
  k_grad<<<(int)(NTOT / 256), 256, 0, stream>>>(image, bre, bim);
  k_conv1_pool<<<(16 * 32 * 256 * 256) / 256, 256, 0, stream>>>(bre, bim, c1w, c1b, pool1);
  k_conv2_avg<<<256, 256, 0, stream>>>(pool1, c2w, c2b, out_spatial, combined);
  // 5) fuse MLP (WMMA) + LayerNorm
  k_fc1<<<32, 32, 0, stream>>>(combined, f1w, f1b, z1);
  k_fc2_ln<<<1, 256, 0, stream>>>(z1, f2w, f2b, lng, lnb, out_holo);
}